// RGCNLayer_48215302865256
// MI455X (gfx1250) — compile-verified
//
#include <hip/hip_runtime.h>
#include <hip/hip_bf16.h>

typedef __attribute__((ext_vector_type(2))) float v2f;
typedef __attribute__((ext_vector_type(8))) float v8f;

#define FLAG_SELF 1   // add x @ loop_weight + bias, write (no accumulate read)
#define FLAG_ACC  2   // accumulate into existing out
#define FLAG_RELU 4   // apply relu before final store

#define LDS_PITCH 132 // 128 + 4 pad: row stride mod 64 banks = 4 -> conflict-free
                      // 132 floats = 528 B, multiple of 16 -> float4/b64 aligned

// ---------------------------------------------------------------------------
// Kernel 1: build fragment-packed weights for all 9 matrices.
//   e in [0,8): W[e] = sum_b coeffs[e,b]*bases[b];  e == 8: loop_weight.
// Packed layout per matrix (16384 floats):
//   Wp[((g*2 + half)*128 + col)*2 + p] = W[k][col],  k = g*4 + half*2 + p
// so a wave's B fragment (v2f per lane) is ONE coalesced global_load_b64:
//   lanes 0-15 (half=0) read cols n0..n0+15 contiguously, lanes 16-31 likewise.
// ---------------------------------------------------------------------------
__global__ void rgcn_wpack(const float* __restrict__ bases,
                           const float* __restrict__ coeffs,
                           const float* __restrict__ loop_w,
                           float* __restrict__ Wp) {
  int o = blockIdx.x * blockDim.x + threadIdx.x;
  if (o >= 9 * 16384) return;
  int e   = o >> 14;
  int r   = o & 16383;
  int p   = r & 1;
  int col = (r >> 1) & 127;
  int hh  = (r >> 8) & 1;
  int g   = r >> 9;
  int k   = g * 4 + hh * 2 + p;
  int si  = k * 128 + col;
  float val;
  if (e < 8) {
    val = 0.0f;
#pragma unroll
    for (int b = 0; b < 4; ++b)
      val += coeffs[e * 4 + b] * bases[b * 16384 + si];
  } else {
    val = loop_w[si];
  }
  Wp[o] = val;
}

// ---------------------------------------------------------------------------
// Kernel 2: grid-stride zero fill (agg + deg region)
// ---------------------------------------------------------------------------
__global__ void rgcn_zero(float* __restrict__ p, size_t n) {
  size_t i = (size_t)blockIdx.x * blockDim.x + threadIdx.x;
  size_t stride = (size_t)gridDim.x * blockDim.x;
  for (; i < n; i += stride) p[i] = 0.0f;
}

// ---------------------------------------------------------------------------
// Kernel 3: scatter-add one etype: agg[dst] += x[src]; deg[dst] += 1
// One wave32 per edge; each lane moves a float4 (32*4 = 128 = D).
// Per-etype working set (x + agg) is L2-resident on MI455X (192 MB L2).
// ---------------------------------------------------------------------------
__global__ __launch_bounds__(256)
void rgcn_scatter(const float* __restrict__ x,
                  const int* __restrict__ src,
                  const int* __restrict__ dst,
                  float* __restrict__ agg,
                  float* __restrict__ deg,
                  int E) {
  int tid  = blockIdx.x * blockDim.x + threadIdx.x;
  int edge = tid >> 5;
  int lane = tid & 31;
  if (edge >= E) return;
  int s = src[edge];
  int d = dst[edge];
  const float4 v = *reinterpret_cast<const float4*>(x + (size_t)s * 128 + lane * 4);
  float* dp = agg + (size_t)d * 128 + lane * 4;
  unsafeAtomicAdd(dp + 0, v.x);
  unsafeAtomicAdd(dp + 1, v.y);
  unsafeAtomicAdd(dp + 2, v.z);
  unsafeAtomicAdd(dp + 3, v.w);
  if (lane == 0) unsafeAtomicAdd(deg + d, 1.0f);
}

// ---------------------------------------------------------------------------
// Kernel 4: out[32-row tile] (+)= (agg/deg) @ W_e [ + x @ W_loop + bias ][relu]
// Block = 256 threads = 8 wave32s; block owns 32 rows (two 16-row M-tiles),
// wave w owns 16-col N-tile n0 = 16*w. A tile staged in LDS with 1/deg scale
// applied at staging (per-row scalar) -> inner loop is
//   ds_load_b64 (A0) + ds_load_b64 (A1) + global_load_b64 (packed B)
//   + 2x v_wmma_f32_16x16x4_f32  (B fragment reused for both M-tiles).
// EXEC stays all-ones: flags are uniform, N divisible by 32.
// ---------------------------------------------------------------------------
__global__ __launch_bounds__(256)
void rgcn_gemm(const float* __restrict__ A,    // agg  [N,128]
               const float* __restrict__ deg,  // [N]
               const float* __restrict__ Wp,   // packed [16384]
               const float* __restrict__ x,    // [N,128] self-loop input
               const float* __restrict__ Wlp,  // packed loop weight [16384]
               const float* __restrict__ bias, // [128]
               float* __restrict__ out,        // [N,128]
               int flags) {
  __shared__ float smem[32 * LDS_PITCH];

  const int tid  = threadIdx.x;
  const int wave = tid >> 5;
  const int lane = tid & 31;
  const int half = lane >> 4;      // 0: K=+0,+1   1: K=+2,+3
  const int lr   = lane & 15;
  const int r0   = blockIdx.x * 32;
  const int col  = wave * 16 + lr;

  // ---- stage scaled A tile (32 x 128) into LDS, coalesced float4 ----
#pragma unroll
  for (int i = 0; i < 4; ++i) {
    int f4  = i * 256 + tid;         // float4 index, 1024 total
    int row = f4 >> 5;               // 32 float4 per row
    int c4  = (f4 & 31) * 4;
    float s = 1.0f / fmaxf(deg[r0 + row], 1.0f);
    float4 v = *reinterpret_cast<const float4*>(A + (size_t)(r0 + row) * 128 + c4);
    v.x *= s; v.y *= s; v.z *= s; v.w *= s;
    *reinterpret_cast<float4*>(&smem[row * LDS_PITCH + c4]) = v;
  }
  __syncthreads();

  v8f c0 = {};
  v8f c1 = {};
  if (flags & FLAG_ACC) {
#pragma unroll
    for (int v = 0; v < 8; ++v) {
      c0[v] = out[(size_t)(r0 + v + 8 * half) * 128 + col];
      c1[v] = out[(size_t)(r0 + 16 + v + 8 * half) * 128 + col];
    }
  }

  const float* bptr = Wp + (size_t)(half * 128 + col) * 2;
  const float* a0   = &smem[lr * LDS_PITCH + 2 * half];
  const float* a1   = a0 + 16 * LDS_PITCH;

#pragma unroll 4
  for (int g = 0; g < 32; ++g) {
    v2f b  = *reinterpret_cast<const v2f*>(bptr + g * 512);
    v2f A0 = *reinterpret_cast<const v2f*>(a0 + g * 4);
    v2f A1 = *reinterpret_cast<const v2f*>(a1 + g * 4);
    c0 = __builtin_amdgcn_wmma_f32_16x16x4_f32(false, A0, false, b,
                                               (short)0, c0, false, false);
    c1 = __builtin_amdgcn_wmma_f32_16x16x4_f32(false, A1, false, b,
                                               (short)0, c1, false, false);
  }

  if (flags & FLAG_SELF) {
    __syncthreads();
    // ---- restage x tile (unscaled) ----
#pragma unroll
    for (int i = 0; i < 4; ++i) {
      int f4  = i * 256 + tid;
      int row = f4 >> 5;
      int c4  = (f4 & 31) * 4;
      float4 v = *reinterpret_cast<const float4*>(x + (size_t)(r0 + row) * 128 + c4);
      *reinterpret_cast<float4*>(&smem[row * LDS_PITCH + c4]) = v;
    }
    __syncthreads();

    const float* blptr = Wlp + (size_t)(half * 128 + col) * 2;
#pragma unroll 4
    for (int g = 0; g < 32; ++g) {
      v2f b  = *reinterpret_cast<const v2f*>(blptr + g * 512);
      v2f A0 = *reinterpret_cast<const v2f*>(a0 + g * 4);
      v2f A1 = *reinterpret_cast<const v2f*>(a1 + g * 4);
      c0 = __builtin_amdgcn_wmma_f32_16x16x4_f32(false, A0, false, b,
                                                 (short)0, c0, false, false);
      c1 = __builtin_amdgcn_wmma_f32_16x16x4_f32(false, A1, false, b,
                                                 (short)0, c1, false, false);
    }
    const float bb = bias[col];
#pragma unroll
    for (int v = 0; v < 8; ++v) { c0[v] += bb; c1[v] += bb; }
  }

#pragma unroll
  for (int v = 0; v < 8; ++v) {
    float v0 = c0[v];
    float v1 = c1[v];
    if (flags & FLAG_RELU) { v0 = fmaxf(v0, 0.0f); v1 = fmaxf(v1, 0.0f); }
    out[(size_t)(r0 + v + 8 * half) * 128 + col]      = v0;
    out[(size_t)(r0 + 16 + v + 8 * half) * 128 + col] = v1;
  }
}

// ---------------------------------------------------------------------------
// Host-side orchestration (graph-capture safe: only kernel launches on stream)
// ---------------------------------------------------------------------------
extern "C" void kernel_launch(void* const* d_in, const int* in_sizes, int n_in,
                              void* d_out, int out_size, void* d_ws, size_t ws_size,
                              hipStream_t stream) {
  const float* x      = (const float*)d_in[0];  // [N,128]
  const int*   src    = (const int*)d_in[1];    // [8,E]
  const int*   dst    = (const int*)d_in[2];    // [8,E]
  const float* bases  = (const float*)d_in[3];  // [4,128,128]
  const float* coeffs = (const float*)d_in[4];  // [8,4]
  const float* loop_w = (const float*)d_in[5];  // [128,128]
  const float* h_bias = (const float*)d_in[6];  // [128]
  float*       out    = (float*)d_out;          // [N,128]

  const int N = in_sizes[0] / 128;              // 100000 (divisible by 32)
  const int E = in_sizes[1] / 8;                // 100000 edges per etype

  // workspace layout (floats): Wp[9*16384] | agg[N*128] | deg[N]  (~52.2 MB)
  float* ws  = (float*)d_ws;
  float* Wp  = ws;
  float* agg = ws + 9 * 16384;
  float* deg = agg + (size_t)N * 128;
  const size_t zcount = (size_t)N * 128 + (size_t)N;  // agg + deg contiguous

  rgcn_wpack<<<(9 * 16384 + 255) / 256, 256, 0, stream>>>(bases, coeffs, loop_w, Wp);

  const int scatterBlocks = (E * 32 + 255) / 256;
  const int gemmBlocks    = N / 32;

  for (int e = 0; e < 8; ++e) {
    rgcn_zero<<<2048, 256, 0, stream>>>(agg, zcount);
    rgcn_scatter<<<scatterBlocks, 256, 0, stream>>>(
        x, src + (size_t)e * E, dst + (size_t)e * E, agg, deg, E);
    int flags = (e == 0) ? FLAG_SELF : FLAG_ACC;
    if (e == 7) flags |= FLAG_RELU;
    rgcn_gemm<<<gemmBlocks, 256, 0, stream>>>(
        agg, deg, Wp + (size_t)e * 16384, x, Wp + (size_t)8 * 16384, h_bias, out, flags);
  }
}